// GraphLossRes_59931973648736
// MI455X (gfx1250) — compile-verified
//
#include <hip/hip_runtime.h>

// CDNA5 / gfx1250: wave32, fp32 WMMA 16x16x4.
typedef float v2f __attribute__((ext_vector_type(2)));
typedef float v8f __attribute__((ext_vector_type(8)));

#define DIMV 32
#define KMATS 4
#define BLK 256
#define NB_MAX 1024

// D = A(16x4) * B(4x16) + C, fp32. One 32x16 product per k needs 2 row-tiles x 8 K-chunks.
__device__ __forceinline__ v8f wmma_f32(v2f a, v2f b, v8f c) {
    return __builtin_amdgcn_wmma_f32_16x16x4_f32(false, a, false, b, (short)0, c, false, false);
}

__global__ __launch_bounds__(BLK) void graph_loss_kernel(
    const float* __restrict__ vertices,   // (N, 32)
    const int*   __restrict__ edges,      // (2, E) flat: [0..E) = src, [E..2E) = dst
    const float* __restrict__ efeat,      // (E,)
    const float* __restrict__ mats,       // (4, 32, 32)
    float*       __restrict__ ws,         // per-block partials
    int E)
{
    __shared__ __align__(16) float smat[KMATS * DIMV * DIMV];  // 16 KB
    __shared__ float sred[BLK];

    const int tid = threadIdx.x;
    for (int i = tid; i < KMATS * DIMV * DIMV; i += BLK) smat[i] = mats[i];
    __syncthreads();

    const int lane   = tid & 31;
    const int el     = lane & 15;   // edge slot within 16-edge tile
    const int h      = lane >> 4;   // half selector (K / row-half split)
    const int gwave  = blockIdx.x * (BLK >> 5) + (tid >> 5);
    const int nwaves = gridDim.x * (BLK >> 5);
    const int ntiles = (E + 15) >> 4;

    float local = 0.0f;

    for (int tile = gwave; tile < ntiles; tile += nwaves) {
        const int  eidx  = tile * 16 + el;
        const bool valid = (eidx < E);
        const int  ec    = valid ? eidx : (E - 1);

        const int si = edges[ec];        // src -> v1
        const int di = edges[E + ec];    // dst -> v2
        const float* v1row = vertices + (long long)si * DIMV;
        const float* v2row = vertices + (long long)di * DIMV;

        // B fragments: V2^T (32 x 16 edges), 8 chunks of K=4.
        // Lane l supplies col (l&15), K rows 4c+2h+{0,1}.
        v2f Bf[8];
        #pragma unroll
        for (int c = 0; c < 8; ++c)
            Bf[c] = *(const v2f*)(v2row + 4 * c + 2 * h);

        // V1 in the accumulator-matching layout: lane holds dims [8h..8h+7] (+16 for row-tile 1).
        const float4 va0 = *(const float4*)(v1row + 8 * h);
        const float4 va1 = *(const float4*)(v1row + 8 * h + 4);
        const float4 vb0 = *(const float4*)(v1row + 8 * h + 16);
        const float4 vb1 = *(const float4*)(v1row + 8 * h + 20);

        const float ef = valid ? efeat[eidx] : 0.0f;

        #pragma unroll
        for (int k = 0; k < KMATS; ++k) {
            const float* mk = &smat[k * DIMV * DIMV];
            v8f acc0 = {};   // rows i = 0..15  of M_k @ v2
            v8f acc1 = {};   // rows i = 16..31
            #pragma unroll
            for (int c = 0; c < 8; ++c) {
                // A fragment: lane l holds row (l&15)[+16], cols 4c+2h+{0,1}
                v2f fa0 = *(const v2f*)(mk + el * DIMV + 4 * c + 2 * h);
                v2f fa1 = *(const v2f*)(mk + (el + 16) * DIMV + 4 * c + 2 * h);
                acc0 = wmma_f32(fa0, Bf[c], acc0);
                acc1 = wmma_f32(fa1, Bf[c], acc1);
            }
            // bilinear[k,e] = sum_i v1[e,i] * (M_k v2)[i]; 16 in-lane FMAs + cross-half add
            float p = acc0[0] * va0.x + acc0[1] * va0.y + acc0[2] * va0.z + acc0[3] * va0.w
                    + acc0[4] * va1.x + acc0[5] * va1.y + acc0[6] * va1.z + acc0[7] * va1.w
                    + acc1[0] * vb0.x + acc1[1] * vb0.y + acc1[2] * vb0.z + acc1[3] * vb0.w
                    + acc1[4] * vb1.x + acc1[5] * vb1.y + acc1[6] * vb1.z + acc1[7] * vb1.w;
            p += __shfl_xor(p, 16, 32);                 // both halves now hold full dot
            const float contrib = ef / (p * p + 1e-4f); // edge_features * 1/(b^2 + eps)
            if (lane < 16) local += contrib;            // count each edge once
        }
    }

    // deterministic block reduction
    sred[tid] = local;
    __syncthreads();
    for (int s = BLK / 2; s > 0; s >>= 1) {
        if (tid < s) sred[tid] += sred[tid + s];
        __syncthreads();
    }
    if (tid == 0) ws[blockIdx.x] = sred[0];
}

__global__ __launch_bounds__(BLK) void final_reduce_kernel(
    const float* __restrict__ ws, int n, float* __restrict__ out)
{
    __shared__ float sred[BLK];
    float s = 0.0f;
    for (int i = threadIdx.x; i < n; i += BLK) s += ws[i];
    sred[threadIdx.x] = s;
    __syncthreads();
    for (int k = BLK / 2; k > 0; k >>= 1) {
        if (threadIdx.x < k) sred[threadIdx.x] += sred[threadIdx.x + k];
        __syncthreads();
    }
    if (threadIdx.x == 0) out[0] = -sred[0];   // negated sum per reference
}

extern "C" void kernel_launch(void* const* d_in, const int* in_sizes, int n_in,
                              void* d_out, int out_size, void* d_ws, size_t ws_size,
                              hipStream_t stream) {
    const float* vertices = (const float*)d_in[0];
    const int*   edges    = (const int*)d_in[1];
    const float* efeat    = (const float*)d_in[2];
    const float* mats     = (const float*)d_in[3];
    const int E = in_sizes[2];   // edge_features element count

    int nb = NB_MAX;
    if (ws_size < (size_t)nb * sizeof(float)) nb = (int)(ws_size / sizeof(float));
    if (nb < 1) nb = 1;

    graph_loss_kernel<<<nb, BLK, 0, stream>>>(vertices, edges, efeat, mats,
                                              (float*)d_ws, E);
    final_reduce_kernel<<<1, BLK, 0, stream>>>((const float*)d_ws, nb, (float*)d_out);
}